// UniGCNConv_82128364634684
// MI455X (gfx1250) — compile-verified
//
#include <hip/hip_runtime.h>
#include <math.h>

typedef float v2f __attribute__((ext_vector_type(2)));
typedef float v4f __attribute__((ext_vector_type(4)));
typedef float v8f __attribute__((ext_vector_type(8)));

#define D_CH 128       // HEADS * OUT_CH == IN_CH == 128
#define ROWS_PB 128    // rows of X per block (amortizes W staging 8x)

// ---------------------------------------------------------------------------
// Kernel 1: Xp = X @ W using V_WMMA_F32_16X16X4_F32 (fp32 matrix pipe).
// Block = 256 threads = 8 wave32 waves; covers 128 rows x 128 cols.
// LDS: W (64KB) + 128x128 A tile (64KB) = 128KB of the 320KB WGP pool.
// Wave w owns row strip [16w,16w+16); loops over 8 column tiles (256 WMMAs).
// ---------------------------------------------------------------------------
__global__ __launch_bounds__(256) void gemm_xw_wmma(
    const float* __restrict__ X, const float* __restrict__ W,
    float* __restrict__ Xp, int N)
{
    __shared__ float sW[D_CH * D_CH];    // 64 KB
    __shared__ float sA[ROWS_PB * D_CH]; // 64 KB

    const int tid  = threadIdx.x;
    const int row0 = blockIdx.x * ROWS_PB;

    // Stage W (128x128 f32) into LDS, 16B per thread per iter.
    for (int i = tid; i < D_CH * D_CH / 4; i += 256) {
        ((v4f*)sW)[i] = ((const v4f*)W)[i];
    }
    // Stage the 128x128 A tile into LDS (guard tail rows).
    for (int i = tid; i < ROWS_PB * D_CH / 4; i += 256) {
        const int r    = i / (D_CH / 4);
        const int c4   = i % (D_CH / 4);
        const int grow = row0 + r;
        v4f v = {0.f, 0.f, 0.f, 0.f};
        if (grow < N) v = ((const v4f*)(X + (size_t)grow * D_CH))[c4];
        ((v4f*)sA)[i] = v;
    }
    __syncthreads();

    const int wave = tid >> 5;
    const int lane = tid & 31;
    const int m    = lane & 15;   // row (A) / col (B,C,D) within 16
    const int kh   = lane >> 4;   // half-select: K offset 0 or 2; D row +0/+8
    const int mrow = wave * 16;   // this wave's row strip within the A tile

    for (int nt = 0; nt < D_CH / 16; ++nt) {
        const int n0 = nt * 16;
        v8f acc = {};
#pragma unroll
        for (int k0 = 0; k0 < D_CH; k0 += 4) {
            const int ka = k0 + 2 * kh;
            v2f a, b;
            // A 16x4 f32 fragment: lane m holds (M=m, K=ka), (M=m, K=ka+1)
            a.x = sA[(mrow + m) * D_CH + ka];
            a.y = sA[(mrow + m) * D_CH + ka + 1];
            // B 4x16 f32 fragment: lane holds (K=ka, N=n0+m), (K=ka+1, N=n0+m)
            b.x = sW[ka       * D_CH + n0 + m];
            b.y = sW[(ka + 1) * D_CH + n0 + m];
            // 8 args: (neg_a, A, neg_b, B, c_mod, C, reuse_a, reuse_b)
            acc = __builtin_amdgcn_wmma_f32_16x16x4_f32(
                false, a, false, b, (short)0, acc, false, false);
        }
        // D 16x16 f32: vgpr i -> row = row0 + mrow + i + 8*kh, col = n0 + m
#pragma unroll
        for (int i = 0; i < 8; ++i) {
            const int r = row0 + mrow + i + 8 * kh;
            if (r < N) Xp[(size_t)r * D_CH + n0 + m] = acc[i];
        }
    }
}

// ---------------------------------------------------------------------------
// Kernel 2: vertex -> edge scatter-add. Two nnz entries per 256-thread block;
// 128 threads cover one entry's 128 channels with f32 atomics (edge_sum is
// 25.6 MB, resident in the 192 MB L2). Thread 0 of each entry bumps the count.
// ---------------------------------------------------------------------------
__global__ __launch_bounds__(256) void scatter_v2e(
    const float* __restrict__ Xp, const int* __restrict__ vertex,
    const int* __restrict__ edges, float* __restrict__ edge_sum,
    float* __restrict__ counts, int nnz)
{
    const int entry = blockIdx.x * 2 + (threadIdx.x >> 7);
    if (entry >= nnz) return;
    const int t = threadIdx.x & 127;
    const int v = vertex[entry];
    const int e = edges[entry];
    atomicAdd(&edge_sum[(size_t)e * D_CH + t], Xp[(size_t)v * D_CH + t]);
    if (t == 0) atomicAdd(&counts[e], 1.0f);
}

// ---------------------------------------------------------------------------
// Kernel 3: Xe = edge_sum / max(count,1) * degE  (in place)
// ---------------------------------------------------------------------------
__global__ __launch_bounds__(256) void edge_normalize(
    float* __restrict__ edge_sum, const float* __restrict__ counts,
    const float* __restrict__ degE, int E)
{
    const int idx = blockIdx.x * 256 + threadIdx.x;
    if (idx >= E * D_CH) return;
    const int e = idx >> 7;
    const float s = degE[e] / fmaxf(counts[e], 1.0f);
    edge_sum[idx] *= s;
}

// ---------------------------------------------------------------------------
// Kernel 4: edge -> vertex scatter-add into zeroed d_out (51 MB, L2-resident).
// ---------------------------------------------------------------------------
__global__ __launch_bounds__(256) void scatter_e2v(
    const float* __restrict__ Xe, const int* __restrict__ vertex,
    const int* __restrict__ edges, float* __restrict__ Xv, int nnz)
{
    const int entry = blockIdx.x * 2 + (threadIdx.x >> 7);
    if (entry >= nnz) return;
    const int t = threadIdx.x & 127;
    const int v = vertex[entry];
    const int e = edges[entry];
    atomicAdd(&Xv[(size_t)v * D_CH + t], Xe[(size_t)e * D_CH + t]);
}

// ---------------------------------------------------------------------------
// Kernel 5: out = gelu_exact(Xv * degV)   (in place on d_out)
// ---------------------------------------------------------------------------
__global__ __launch_bounds__(256) void finalize_gelu(
    float* __restrict__ Xv, const float* __restrict__ degV, int N)
{
    const int idx = blockIdx.x * 256 + threadIdx.x;
    if (idx >= N * D_CH) return;
    const float x = Xv[idx] * degV[idx >> 7];
    // exact (erf) GELU: 0.5 * x * (1 + erf(x / sqrt(2)))
    Xv[idx] = 0.5f * x * (1.0f + erff(x * 0.70710678118654752440f));
}

// ---------------------------------------------------------------------------
// Host-side launcher (graph-capture safe: only async ops on `stream`).
// Inputs (setup_inputs order): X, W, degE, degV, vertex, edges.
// ---------------------------------------------------------------------------
extern "C" void kernel_launch(void* const* d_in, const int* in_sizes, int n_in,
                              void* d_out, int out_size, void* d_ws, size_t ws_size,
                              hipStream_t stream) {
    const float* X      = (const float*)d_in[0];
    const float* W      = (const float*)d_in[1];
    const float* degE   = (const float*)d_in[2];
    const float* degV   = (const float*)d_in[3];
    const int*   vertex = (const int*)d_in[4];
    const int*   edges  = (const int*)d_in[5];

    const int N   = in_sizes[0] / D_CH;  // 100000
    const int E   = in_sizes[2];         // 50000
    const int NNZ = in_sizes[4];         // 1600000

    // Workspace layout: Xp [N*128] | edge_sum [E*128] | counts [E]
    float* Xp       = (float*)d_ws;
    float* edge_sum = Xp + (size_t)N * D_CH;
    float* counts   = edge_sum + (size_t)E * D_CH;
    float* out      = (float*)d_out;

    // Zero the accumulators (capture-safe async memsets).
    hipMemsetAsync(edge_sum, 0, ((size_t)E * D_CH + (size_t)E) * sizeof(float), stream);
    hipMemsetAsync(out,      0, (size_t)N * D_CH * sizeof(float), stream);

    // 1) Xp = X @ W on the WMMA pipe (fp32).
    gemm_xw_wmma<<<(N + ROWS_PB - 1) / ROWS_PB, 256, 0, stream>>>(X, W, Xp, N);

    // 2) vertex -> edge segment-sum + counts.
    scatter_v2e<<<(NNZ + 1) / 2, 256, 0, stream>>>(Xp, vertex, edges, edge_sum, counts, NNZ);

    // 3) mean + degE scaling, in place.
    edge_normalize<<<(E * D_CH + 255) / 256, 256, 0, stream>>>(edge_sum, counts, degE, E);

    // 4) edge -> vertex segment-sum into d_out.
    scatter_e2v<<<(NNZ + 1) / 2, 256, 0, stream>>>(edge_sum, vertex, edges, out, NNZ);

    // 5) degV scaling + exact GELU, in place on d_out.
    finalize_gelu<<<(N * D_CH + 255) / 256, 256, 0, stream>>>(out, degV, N);
}